// GraphExpert_18631568130677
// MI455X (gfx1250) — compile-verified
//
#include <hip/hip_runtime.h>
#include <math.h>

typedef __attribute__((ext_vector_type(2))) float v2f;
typedef __attribute__((ext_vector_type(8))) float v8f;

#define RELS 2

// ---------------------------------------------------------------------------
// W[r] = comp[r,0]*basis[0] + comp[r,1]*basis[1]   (n = in*out elements each)
// ---------------------------------------------------------------------------
__global__ void combine_basis(const float* __restrict__ comp,
                              const float* __restrict__ basis,
                              float* __restrict__ W, int n) {
    int i = blockIdx.x * blockDim.x + threadIdx.x;
    if (i >= RELS * n) return;
    int r = i / n, e = i % n;
    W[i] = comp[r * RELS + 0] * basis[e] + comp[r * RELS + 1] * basis[n + e];
}

// ---------------------------------------------------------------------------
// per-(dst,rel) in-degree count, then invert
// ---------------------------------------------------------------------------
__global__ void count_edges(const int* __restrict__ dst,
                            const int* __restrict__ et,
                            float* __restrict__ cnt, int E) {
    int i = blockIdx.x * blockDim.x + threadIdx.x;
    if (i >= E) return;
    unsafeAtomicAdd(&cnt[dst[i] * RELS + et[i]], 1.0f);
}

__global__ void invert_cnt(float* __restrict__ c, int n) {
    int i = blockIdx.x * blockDim.x + threadIdx.x;
    if (i < n) c[i] = 1.0f / fmaxf(c[i], 1.0f);
}

// ---------------------------------------------------------------------------
// C[z] = A[M,128] @ B[z][128,NOUT]  (+Cin, +bias, relu per flags)
// grid: (ceil(M/128), 1, nbatch), block 256 = 8 waves.
// Each wave computes a full 16-row x NOUT-col strip: NOUT/16 accumulators,
// each A fragment is loaded once and reused for NOUT/16 WMMAs.
// Full B panel staged in LDS with row stride NOUT+16 so the 4 K-rows touched
// by a B-fragment quad map to bank groups {c, c+16, c+32, c+48}: conflict-free.
// flags: 1 = add Cin, 2 = add bias, 4 = relu
// ---------------------------------------------------------------------------
template <int NOUT>
__global__ __launch_bounds__(256) void wmma_gemm_k128(
    const float* __restrict__ A, const float* __restrict__ B,
    const float* __restrict__ bias, const float* __restrict__ Cin,
    float* __restrict__ Cout, int M,
    long strideB, long strideC, int flags)
{
    constexpr int NT = NOUT / 16;   // 16-col tiles per strip
    constexpr int NS = NOUT + 16;   // padded LDS row stride (floats)
    __shared__ float lB[128 * NS];

    const int z = blockIdx.z;
    const float* Bz = B + (long)z * strideB;

    // cooperative load B[128][NOUT] -> LDS (float4, padded rows)
    for (int i = threadIdx.x * 4; i < 128 * NOUT; i += 256 * 4) {
        int k = i / NOUT, c = i % NOUT;
        float4 t = *(const float4*)(Bz + i);
        *(float4*)(&lB[k * NS + c]) = t;
    }
    __syncthreads();

    const int wave  = threadIdx.x >> 5;
    const int lane  = threadIdx.x & 31;
    const int tileM = blockIdx.x * 8 + wave;
    if (tileM * 16 >= M) return;           // wave-uniform: EXEC stays all-ones

    const int row = tileM * 16 + (lane & 15);  // A-frag: lane holds row
    const int kb  = (lane >> 4) << 1;          // lanes 0-15: K+0/1, 16-31: K+2/3
    const int col = lane & 15;

    const float* Arow = A + (long)row * 128;
    v8f acc[NT];
    #pragma unroll
    for (int t = 0; t < NT; ++t) acc[t] = (v8f){};

    #pragma unroll
    for (int k0 = 0; k0 < 128; k0 += 4) {
        v2f a = *(const v2f*)(Arow + k0 + kb);
        #pragma unroll
        for (int t = 0; t < NT; ++t) {
            v2f b;
            b.x = lB[(k0 + kb)     * NS + t * 16 + col];
            b.y = lB[(k0 + kb + 1) * NS + t * 16 + col];
            acc[t] = __builtin_amdgcn_wmma_f32_16x16x4_f32(
                false, a, false, b, (short)0, acc[t], false, false);
        }
    }

    // C/D layout: acc[t][i] = C[tile16 + i + (lane>=16 ? 8 : 0)][t*16 + col]
    float* Cz = Cout + (long)z * strideC;
    const float* Ciz = Cin + (long)z * strideC;   // only read if flags&1
    const int mbase = tileM * 16 + ((lane >> 4) << 3);
    #pragma unroll
    for (int t = 0; t < NT; ++t) {
        #pragma unroll
        for (int i = 0; i < 8; ++i) {
            long idx = (long)(mbase + i) * NOUT + t * 16 + col;
            float v = acc[t][i];
            if (flags & 1) v += Ciz[idx];
            if (flags & 2) v += bias[t * 16 + col];
            if (flags & 4) v = fmaxf(v, 0.0f);
            Cz[idx] = v;
        }
    }
}

// ---------------------------------------------------------------------------
// per-edge mean-scatter: out[dst] += h[rel][src] * invdeg[dst,rel]
// one lane handles 4 output dims of one edge
// ---------------------------------------------------------------------------
template <int D>
__global__ __launch_bounds__(256) void scatter_edges(
    const float* __restrict__ h, long hRelStride,
    const int* __restrict__ src, const int* __restrict__ dst,
    const int* __restrict__ et, const float* __restrict__ invc,
    float* __restrict__ out, int E)
{
    const int lpe = D / 4;
    long gid = (long)blockIdx.x * blockDim.x + threadIdx.x;
    long edge = gid / lpe;
    if (edge >= E) return;
    int j = (int)(gid % lpe) * 4;
    int r = et[edge], s = src[edge], d = dst[edge];
    float w = invc[(long)d * RELS + r];
    float4 v = *(const float4*)(h + (long)r * hRelStride + (long)s * D + j);
    float* o = out + (long)d * D + j;
    unsafeAtomicAdd(o + 0, v.x * w);
    unsafeAtomicAdd(o + 1, v.y * w);
    unsafeAtomicAdd(o + 2, v.z * w);
    unsafeAtomicAdd(o + 3, v.w * w);
}

// ---------------------------------------------------------------------------
// gather 4096 nodes, run 64->64(LN,relu)->32(LN,relu)->1 sigmoid head
// one thread per node (tiny workload)
// ---------------------------------------------------------------------------
__global__ void gather_mlp(const float* __restrict__ y1,
                           const int* __restrict__ nidx,
                           const float* __restrict__ mW1, const float* __restrict__ mb1,
                           const float* __restrict__ mg1, const float* __restrict__ mbe1,
                           const float* __restrict__ mW2, const float* __restrict__ mb2,
                           const float* __restrict__ mg2, const float* __restrict__ mbe2,
                           const float* __restrict__ mW3, const float* __restrict__ mb3,
                           float* __restrict__ out_e, float* __restrict__ out_p, int B)
{
    int i = blockIdx.x * blockDim.x + threadIdx.x;
    if (i >= B) return;
    int n = nidx[i];

    float e[64];
    for (int j = 0; j < 64; ++j) {
        e[j] = y1[(long)n * 64 + j];
        out_e[(long)i * 64 + j] = e[j];
    }

    float h[64];
    for (int k = 0; k < 64; ++k) {
        float s = mb1[k];
        for (int j = 0; j < 64; ++j) s += e[j] * mW1[j * 64 + k];
        h[k] = s;
    }
    float m = 0.f;
    for (int k = 0; k < 64; ++k) m += h[k];
    m *= (1.0f / 64.0f);
    float var = 0.f;
    for (int k = 0; k < 64; ++k) { float d = h[k] - m; var += d * d; }
    var *= (1.0f / 64.0f);
    float is = rsqrtf(var + 1e-5f);
    for (int k = 0; k < 64; ++k)
        h[k] = fmaxf(mg1[k] * (h[k] - m) * is + mbe1[k], 0.0f);

    float h2[32];
    for (int k = 0; k < 32; ++k) {
        float s = mb2[k];
        for (int j = 0; j < 64; ++j) s += h[j] * mW2[j * 32 + k];
        h2[k] = s;
    }
    m = 0.f;
    for (int k = 0; k < 32; ++k) m += h2[k];
    m *= (1.0f / 32.0f);
    var = 0.f;
    for (int k = 0; k < 32; ++k) { float d = h2[k] - m; var += d * d; }
    var *= (1.0f / 32.0f);
    is = rsqrtf(var + 1e-5f);
    for (int k = 0; k < 32; ++k)
        h2[k] = fmaxf(mg2[k] * (h2[k] - m) * is + mbe2[k], 0.0f);

    float zacc = mb3[0];
    for (int j = 0; j < 32; ++j) zacc += h2[j] * mW3[j];
    out_p[i] = 1.0f / (1.0f + expf(-zacc));
}

// ---------------------------------------------------------------------------
extern "C" void kernel_launch(void* const* d_in, const int* in_sizes, int n_in,
                              void* d_out, int out_size, void* d_ws, size_t ws_size,
                              hipStream_t stream) {
    const int*   nidx   = (const int*)d_in[0];
    const int*   ei     = (const int*)d_in[1];
    const int*   et     = (const int*)d_in[2];
    const float* x      = (const float*)d_in[3];
    const float* basis0 = (const float*)d_in[4];
    const float* comp0  = (const float*)d_in[5];
    const float* root0  = (const float*)d_in[6];
    const float* bias0  = (const float*)d_in[7];
    const float* basis1 = (const float*)d_in[8];
    const float* comp1  = (const float*)d_in[9];
    const float* root1  = (const float*)d_in[10];
    const float* bias1  = (const float*)d_in[11];
    const float* mW1 = (const float*)d_in[12];
    const float* mb1 = (const float*)d_in[13];
    const float* mg1 = (const float*)d_in[14];
    const float* mbe1= (const float*)d_in[15];
    const float* mW2 = (const float*)d_in[16];
    const float* mb2 = (const float*)d_in[17];
    const float* mg2 = (const float*)d_in[18];
    const float* mbe2= (const float*)d_in[19];
    const float* mW3 = (const float*)d_in[20];
    const float* mb3 = (const float*)d_in[21];

    const int E  = in_sizes[2];
    const int N  = in_sizes[3] / 128;   // 100000 (multiple of 16)
    const int Bn = in_sizes[0];         // 4096
    const int* srcv = ei;
    const int* dstv = ei + E;

    // workspace layout
    char* ws = (char*)d_ws;
    size_t off = 0;
    float* W0c  = (float*)(ws + off); off += (size_t)RELS * 128 * 128 * 4;
    float* W1c  = (float*)(ws + off); off += (size_t)RELS * 128 * 64 * 4;
    float* invc = (float*)(ws + off); off += (size_t)N * RELS * 4;
    float* hbuf = (float*)(ws + off); off += (size_t)RELS * N * 128 * 4;
    float* y0   = (float*)(ws + off); off += (size_t)N * 128 * 4;
    float* y1   = (float*)(ws + off); off += (size_t)N * 64 * 4;

    hipMemsetAsync(invc, 0, (size_t)N * RELS * 4, stream);
    hipMemsetAsync(y0,   0, (size_t)N * 128 * 4, stream);
    hipMemsetAsync(y1,   0, (size_t)N * 64 * 4, stream);

    // relation weights from bases
    combine_basis<<<(RELS * 128 * 128 + 255) / 256, 256, 0, stream>>>(comp0, basis0, W0c, 128 * 128);
    combine_basis<<<(RELS * 128 * 64  + 255) / 256, 256, 0, stream>>>(comp1, basis1, W1c, 128 * 64);

    // shared in-degree counts (same edges both layers)
    count_edges<<<(E + 255) / 256, 256, 0, stream>>>(dstv, et, invc, E);
    invert_cnt<<<(RELS * N + 255) / 256, 256, 0, stream>>>(invc, RELS * N);

    const int mBlocks = (N / 16 + 7) / 8;

    // ---- layer 0: h = x @ W0[r]  ->  mean-scatter  ->  + x@root0 + b0, relu
    wmma_gemm_k128<128><<<dim3(mBlocks, 1, RELS), 256, 0, stream>>>(
        x, W0c, nullptr, y0, hbuf, N, (long)128 * 128, (long)N * 128, 0);
    {
        long th = (long)E * 32;
        scatter_edges<128><<<(unsigned)((th + 255) / 256), 256, 0, stream>>>(
            hbuf, (long)N * 128, srcv, dstv, et, invc, y0, E);
    }
    wmma_gemm_k128<128><<<dim3(mBlocks, 1, 1), 256, 0, stream>>>(
        x, root0, bias0, y0, y0, N, 0, 0, 1 | 2 | 4);

    // ---- layer 1: h = y0 @ W1[r] ->  mean-scatter  ->  + y0@root1 + b1
    wmma_gemm_k128<64><<<dim3(mBlocks, 1, RELS), 256, 0, stream>>>(
        y0, W1c, nullptr, y1, hbuf, N, (long)128 * 64, (long)N * 64, 0);
    {
        long th = (long)E * 16;
        scatter_edges<64><<<(unsigned)((th + 255) / 256), 256, 0, stream>>>(
            hbuf, (long)N * 64, srcv, dstv, et, invc, y1, E);
    }
    wmma_gemm_k128<64><<<dim3(mBlocks, 1, 1), 256, 0, stream>>>(
        y0, root1, bias1, y1, y1, N, 0, 0, 1 | 2);

    // ---- gather + classifier head; outputs: [Bn*64 expert | Bn probs]
    float* out_e = (float*)d_out;
    float* out_p = out_e + (size_t)Bn * 64;
    gather_mlp<<<(Bn + 63) / 64, 64, 0, stream>>>(y1, nidx,
        mW1, mb1, mg1, mbe1, mW2, mb2, mg2, mbe2, mW3, mb3,
        out_e, out_p, Bn);
}